// GraphSAGELayer_10385230921949
// MI455X (gfx1250) — compile-verified
//
#include <hip/hip_runtime.h>

#define N_NODES 50000
#define N_EDGES 800000
#define D 128
#define LN_EPS 1e-5f
#define WPB 4          // waves (tiles) per block
#define LSTRIDE 132    // padded LDS row stride (floats)

typedef __attribute__((ext_vector_type(2))) float v2f;
typedef __attribute__((ext_vector_type(8))) float v8f;

__global__ void zero_ws_kernel(float* __restrict__ ws, int n) {
    int i = blockIdx.x * blockDim.x + threadIdx.x;
    if (i < n) ws[i] = 0.0f;
}

// One wave32 per edge: scalar row/col, coalesced float4 gather, f32 atomic scatter.
__global__ void scatter_edges_kernel(const float* __restrict__ x,
                                     const int* __restrict__ ei,
                                     float* __restrict__ nbr_sum,
                                     float* __restrict__ deg) {
    int gid  = blockIdx.x * blockDim.x + threadIdx.x;
    int edge = gid >> 5;
    int lane = threadIdx.x & 31;
    if (edge >= N_EDGES) return;
    int row = ei[edge];            // source
    int col = ei[N_EDGES + edge];  // destination
    const float4* xs = (const float4*)(x + (size_t)row * D);
    float4 v = xs[lane];
    float* dst = nbr_sum + (size_t)col * D + lane * 4;
    atomicAdd(dst + 0, v.x);
    atomicAdd(dst + 1, v.y);
    atomicAdd(dst + 2, v.z);
    atomicAdd(dst + 3, v.w);
    if (lane == 0) atomicAdd(deg + col, 1.0f);
}

// One wave per 16-row node tile: GEMM1 (WMMA f32) -> LDS restage -> GEMM2 (WMMA f32)
// -> bias+ReLU -> LayerNorm (shfl_xor reductions) -> store.
__global__ __launch_bounds__(WPB * 32) void sage_fused_kernel(
    const float* __restrict__ x,
    const float* __restrict__ nbr_sum,
    const float* __restrict__ deg,
    const float* __restrict__ agg_W,
    const float* __restrict__ agg_b,
    const float* __restrict__ W,
    const float* __restrict__ bvec,
    const float* __restrict__ gamma,
    const float* __restrict__ beta,
    float* __restrict__ out) {
    __shared__ float lds[WPB * 16 * LSTRIDE];

    const int lane = threadIdx.x & 31;
    const int wave = threadIdx.x >> 5;
    const int tile = blockIdx.x * WPB + wave;
    if (tile * 16 >= N_NODES) return;   // whole-wave exit; active waves keep EXEC all-1s

    const int m0 = tile * 16;
    const int nn = lane & 15;           // m for A-frag, n for B/C-frags
    const int h  = lane >> 4;           // half-wave select
    const int kq = h * 2;               // k sub-offset within a K=4 step
    float* tlds = lds + wave * 16 * LSTRIDE;

    const int arow = m0 + nn;
    const float invdeg = 1.0f / fmaxf(deg[arow], 1.0f);

    v8f acc[8];
    #pragma unroll
    for (int j = 0; j < 8; ++j)
        #pragma unroll
        for (int i = 0; i < 8; ++i) acc[j][i] = 0.0f;

    // ---- GEMM1: agg = (nbr_sum * invdeg) @ agg_W ----
    const float* ap = nbr_sum + (size_t)arow * D;
    for (int k = 0; k < D; k += 4) {
        v2f a;
        a.x = ap[k + kq]     * invdeg;
        a.y = ap[k + kq + 1] * invdeg;
        #pragma unroll
        for (int j = 0; j < 8; ++j) {
            v2f bb;
            bb.x = agg_W[(k + kq)     * D + j * 16 + nn];
            bb.y = agg_W[(k + kq + 1) * D + j * 16 + nn];
            acc[j] = __builtin_amdgcn_wmma_f32_16x16x4_f32(
                false, a, false, bb, (short)0, acc[j], false, false);
        }
    }

    // add agg_b, restage C-layout tile into LDS as a row-major 16x128 A source
    #pragma unroll
    for (int j = 0; j < 8; ++j) {
        float bias = agg_b[j * 16 + nn];
        #pragma unroll
        for (int i = 0; i < 8; ++i)
            tlds[(i + 8 * h) * LSTRIDE + j * 16 + nn] = acc[j][i] + bias;
    }
    // wave-private LDS: compiler inserts the required s_wait_dscnt before reads

    #pragma unroll
    for (int j = 0; j < 8; ++j)
        #pragma unroll
        for (int i = 0; i < 8; ++i) acc[j][i] = 0.0f;

    // ---- GEMM2 part A: x @ W[0:128,:] ----
    const float* xp = x + (size_t)arow * D;
    for (int k = 0; k < D; k += 4) {
        v2f a;
        a.x = xp[k + kq];
        a.y = xp[k + kq + 1];
        #pragma unroll
        for (int j = 0; j < 8; ++j) {
            v2f bb;
            bb.x = W[(k + kq)     * D + j * 16 + nn];
            bb.y = W[(k + kq + 1) * D + j * 16 + nn];
            acc[j] = __builtin_amdgcn_wmma_f32_16x16x4_f32(
                false, a, false, bb, (short)0, acc[j], false, false);
        }
    }
    // ---- GEMM2 part B: agg @ W[128:256,:] ----
    const float* Wlo = W + D * D;
    const float* gp  = tlds + nn * LSTRIDE;
    for (int k = 0; k < D; k += 4) {
        v2f a;
        a.x = gp[k + kq];
        a.y = gp[k + kq + 1];
        #pragma unroll
        for (int j = 0; j < 8; ++j) {
            v2f bb;
            bb.x = Wlo[(k + kq)     * D + j * 16 + nn];
            bb.y = Wlo[(k + kq + 1) * D + j * 16 + nn];
            acc[j] = __builtin_amdgcn_wmma_f32_16x16x4_f32(
                false, a, false, bb, (short)0, acc[j], false, false);
        }
    }

    // ---- bias + ReLU + per-row partial moments ----
    float mu[8], rs[8];
    #pragma unroll
    for (int i = 0; i < 8; ++i) { mu[i] = 0.0f; rs[i] = 0.0f; }
    #pragma unroll
    for (int j = 0; j < 8; ++j) {
        float bias = bvec[j * 16 + nn];
        #pragma unroll
        for (int i = 0; i < 8; ++i) {
            float v = fmaxf(acc[j][i] + bias, 0.0f);
            acc[j][i] = v;
            mu[i] += v;
            rs[i] += v * v;
        }
    }
    // reduce each row's 128 values across its 16-lane half (xor masks stay in-half)
    #pragma unroll
    for (int i = 0; i < 8; ++i) {
        float s = mu[i], ss = rs[i];
        #pragma unroll
        for (int m = 1; m < 16; m <<= 1) {
            s  += __shfl_xor(s,  m, 32);
            ss += __shfl_xor(ss, m, 32);
        }
        float mean = s * (1.0f / D);
        float var  = ss * (1.0f / D) - mean * mean;
        mu[i] = mean;
        rs[i] = rsqrtf(var + LN_EPS);
    }

    // ---- normalize, scale/shift, store ----
    #pragma unroll
    for (int j = 0; j < 8; ++j) {
        float g  = gamma[j * 16 + nn];
        float be = beta[j * 16 + nn];
        #pragma unroll
        for (int i = 0; i < 8; ++i) {
            float o = (acc[j][i] - mu[i]) * rs[i] * g + be;
            out[(size_t)(m0 + i + 8 * h) * D + j * 16 + nn] = o;
        }
    }
}

extern "C" void kernel_launch(void* const* d_in, const int* in_sizes, int n_in,
                              void* d_out, int out_size, void* d_ws, size_t ws_size,
                              hipStream_t stream) {
    const float* x     = (const float*)d_in[0];
    const int*   ei    = (const int*)d_in[1];
    const float* agg_W = (const float*)d_in[2];
    const float* agg_b = (const float*)d_in[3];
    const float* W     = (const float*)d_in[4];
    const float* b     = (const float*)d_in[5];
    const float* gamma = (const float*)d_in[6];
    const float* beta  = (const float*)d_in[7];
    float* out = (float*)d_out;

    float* nbr_sum = (float*)d_ws;                     // N_NODES * D floats
    float* deg     = nbr_sum + (size_t)N_NODES * D;    // N_NODES floats

    int nz = N_NODES * D + N_NODES;
    zero_ws_kernel<<<(nz + 255) / 256, 256, 0, stream>>>((float*)d_ws, nz);

    long long sthreads = (long long)N_EDGES * 32;
    scatter_edges_kernel<<<(int)((sthreads + 255) / 256), 256, 0, stream>>>(
        x, ei, nbr_sum, deg);

    int ntiles = (N_NODES + 15) / 16;   // 3125
    sage_fused_kernel<<<(ntiles + WPB - 1) / WPB, WPB * 32, 0, stream>>>(
        x, nbr_sum, deg, agg_W, agg_b, W, b, gamma, beta, out);
}